// Kernel_6199962355332830965_53472342835845
// MI455X (gfx1250) — compile-verified
//
#include <hip/hip_runtime.h>

// Problem constants (frozen by the reference)
constexpr int N_  = 16;
constexpr int C_  = 128;
constexpr int H_  = 48;
constexpr int W_  = 48;
constexpr int HW_ = H_ * W_;          // 2304
constexpr int KSPLIT = 4;

typedef float v2f __attribute__((ext_vector_type(2)));
typedef float v8f __attribute__((ext_vector_type(8)));

// CDNA5 fp32 WMMA: D(16x16) = A(16x4) * B(4x16) + C
__device__ __forceinline__ v8f wmma4(v2f a, v2f b, v8f c) {
  return __builtin_amdgcn_wmma_f32_16x16x4_f32(
      /*neg_a=*/false, a, /*neg_b=*/false, b,
      /*c_mod=*/(short)0, c, /*reuse_a=*/false, /*reuse_b=*/false);
}

// ---------------------------------------------------------------------------
// Elementwise: t5 = p5_w * max(-x, -roll_w(x,1))
// ---------------------------------------------------------------------------
__global__ void k_t5(const float* __restrict__ x, const float* __restrict__ p5,
                     float* __restrict__ t5) {
  int i = blockIdx.x * blockDim.x + threadIdx.x;
  int total = N_ * C_ * HW_;
  if (i >= total) return;
  int w = i % W_;
  int im = (w == 0) ? (i + W_ - 1) : (i - 1);   // (w-1) mod W
  float v = fmaxf(-x[i], -x[im]);
  t5[i] = p5[i % (C_ * HW_)] * v;
}

// ---------------------------------------------------------------------------
// t1[n,h,w] = max over channels of x
// ---------------------------------------------------------------------------
__global__ void k_t1(const float* __restrict__ x, float* __restrict__ t1) {
  int i = blockIdx.x * blockDim.x + threadIdx.x;
  if (i >= N_ * HW_) return;
  int n = i / HW_, q = i % HW_;
  const float* p = x + (size_t)n * C_ * HW_ + q;
  float m = p[0];
  for (int c = 1; c < C_; ++c) m = fmaxf(m, p[(size_t)c * HW_]);
  t1[i] = m;
}

// ---------------------------------------------------------------------------
// weff[o,k,j] = sum_oc w10[o, oc*3+k] * w8[oc, j]   (128 x 3 x 5)
// ---------------------------------------------------------------------------
__global__ void k_weff(const float* __restrict__ w10, const float* __restrict__ w8,
                       float* __restrict__ weff) {
  int i = blockIdx.x * blockDim.x + threadIdx.x;
  if (i >= C_ * 15) return;
  int o = i / 15, kj = i % 15, k = kj / 5, j = kj % 5;
  float s = 0.f;
  for (int oc = 0; oc < C_ / 2; ++oc)
    s += w10[o * (3 * C_ / 2) + oc * 3 + k] * w8[oc * 5 + j];
  weff[i] = s;
}

// ---------------------------------------------------------------------------
// t12 = max(x, t10[(c-1)%C]) where t10[o] = sum_{k,j} weff[o,k,j]*t1[h+2k-2, w+3j-6]
// ---------------------------------------------------------------------------
__global__ void k_t12(const float* __restrict__ x, const float* __restrict__ t1,
                      const float* __restrict__ weff, float* __restrict__ t12) {
  int i = blockIdx.x * blockDim.x + threadIdx.x;
  int total = N_ * C_ * HW_;
  if (i >= total) return;
  int w = i % W_;
  int h = (i / W_) % H_;
  int c = (i / HW_) % C_;
  int n = i / (C_ * HW_);
  int o = (c - 1 + C_) % C_;                 // channel roll by +1
  const float* wv  = weff + o * 15;
  const float* t1n = t1 + (size_t)n * HW_;
  float s = 0.f;
#pragma unroll
  for (int k = 0; k < 3; ++k) {
    int hh = h + 2 * k - 2;
    if (hh < 0 || hh >= H_) continue;
#pragma unroll
    for (int j = 0; j < 5; ++j) {
      int ww = w + 3 * j - 6;
      if (ww < 0 || ww >= W_) continue;
      s += wv[k * 5 + j] * t1n[hh * W_ + ww];
    }
  }
  t12[i] = fmaxf(x[i], s);
}

// ---------------------------------------------------------------------------
// GEMM NN: D[n] = scale * A[n](MxK, row-major) @ B[n](KxNcols, row-major)
// One wave (32 thr) computes a 16x64 tile; K stepped by 4 via v_wmma_f32_16x16x4_f32.
// grid = (Ncols/64, M/16, batches)
// ---------------------------------------------------------------------------
__global__ __launch_bounds__(32)
void gemm_nn_wmma(const float* __restrict__ A, size_t strideA, int lda,
                  const float* __restrict__ B, size_t strideB, int ldb,
                  float* __restrict__ D, size_t strideD, int ldd,
                  int K, float scale) {
  const int lane = threadIdx.x;         // 0..31
  const int half = lane >> 4;           // 0: lanes 0-15, 1: lanes 16-31
  const int l    = lane & 15;
  const int n    = blockIdx.z;
  const int row0 = blockIdx.y * 16;
  const int col0 = blockIdx.x * 64;

  const float* Ab = A + (size_t)n * strideA + (size_t)(row0 + l) * lda;
  const float* Bb = B + (size_t)n * strideB + col0 + l;
  float*       Db = D + (size_t)n * strideD;

  v8f acc0 = {}, acc1 = {}, acc2 = {}, acc3 = {};
#pragma unroll 4
  for (int k = 0; k < K; k += 4) {
    const int kk = k + 2 * half;
    v2f a;
    a.x = Ab[kk];
    a.y = Ab[kk + 1];
    const float* b0 = Bb + (size_t)kk * ldb;        // row kk
    const float* b1 = b0 + ldb;                     // row kk+1
    v2f bf0; bf0.x = b0[0];  bf0.y = b1[0];
    v2f bf1; bf1.x = b0[16]; bf1.y = b1[16];
    v2f bf2; bf2.x = b0[32]; bf2.y = b1[32];
    v2f bf3; bf3.x = b0[48]; bf3.y = b1[48];
    acc0 = wmma4(a, bf0, acc0);
    acc1 = wmma4(a, bf1, acc1);
    acc2 = wmma4(a, bf2, acc2);
    acc3 = wmma4(a, bf3, acc3);
  }
#pragma unroll
  for (int r = 0; r < 8; ++r) {
    float* drow = Db + (size_t)(row0 + r + 8 * half) * ldd + col0 + l;
    drow[0]  = scale * acc0[r];
    drow[16] = scale * acc1[r];
    drow[32] = scale * acc2[r];
    drow[48] = scale * acc3[r];
  }
}

// ---------------------------------------------------------------------------
// GEMM NT split-K: Dpart[ks][n] = A[n](rows,K) @ B[n](cols,K)^T over K-chunk ks.
// Both operands K-contiguous. One wave computes one 16x16 tile partial.
// grid = (cols/16, rows/16, batches*KSPLIT)
// ---------------------------------------------------------------------------
__global__ __launch_bounds__(32)
void gemm_nt_wmma(const float* __restrict__ A, size_t strideA, int lda,
                  const float* __restrict__ B, size_t strideB, int ldb,
                  float* __restrict__ Dpart, int K) {
  const int lane = threadIdx.x;
  const int half = lane >> 4;
  const int l    = lane & 15;
  const int z    = blockIdx.z;
  const int n    = z / KSPLIT;
  const int ks   = z % KSPLIT;
  const int kchunk = K / KSPLIT;
  const int kbeg   = ks * kchunk;
  const int row0 = blockIdx.y * 16;
  const int col0 = blockIdx.x * 16;

  const float* Ab = A + (size_t)n * strideA + (size_t)(row0 + l) * lda;
  const float* Bb = B + (size_t)n * strideB + (size_t)(col0 + l) * ldb;

  v8f acc = {};
#pragma unroll 8
  for (int k = kbeg; k < kbeg + kchunk; k += 4) {
    const int kk = k + 2 * half;
    v2f a; a.x = Ab[kk]; a.y = Ab[kk + 1];
    v2f b; b.x = Bb[kk]; b.y = Bb[kk + 1];
    acc = wmma4(a, b, acc);
  }
  float* Db = Dpart + ((size_t)ks * N_ + n) * (C_ * C_);
#pragma unroll
  for (int r = 0; r < 8; ++r)
    Db[(size_t)(row0 + r + 8 * half) * C_ + col0 + l] = acc[r];
}

// ---------------------------------------------------------------------------
// Deterministic split-K reduction: M[i] = sum_ks Mpart[ks][i]
// ---------------------------------------------------------------------------
__global__ void k_reduceM(const float* __restrict__ Mpart, float* __restrict__ M) {
  int i = blockIdx.x * blockDim.x + threadIdx.x;
  const int NM = N_ * C_ * C_;
  if (i >= NM) return;
  float s = 0.f;
#pragma unroll
  for (int ks = 0; ks < KSPLIT; ++ks) s += Mpart[(size_t)ks * NM + i];
  M[i] = s;
}

// ---------------------------------------------------------------------------
extern "C" void kernel_launch(void* const* d_in, const int* in_sizes, int n_in,
                              void* d_out, int out_size, void* d_ws, size_t ws_size,
                              hipStream_t stream) {
  const float* x   = (const float*)d_in[0];   // (16,128,48,48)
  const float* p5  = (const float*)d_in[1];   // (1,128,48,48)
  const float* w6  = (const float*)d_in[2];   // (128,128)
  const float* w8  = (const float*)d_in[3];   // (64,1,1,5)
  const float* w10 = (const float*)d_in[4];   // (128,192)
  float* out = (float*)d_out;                 // (16,128,48,48)

  const size_t NCHW = (size_t)N_ * C_ * HW_;  // 4,718,592
  float* ws    = (float*)d_ws;
  float* t5buf = ws;                          // reused as t12 after t6 GEMM
  float* t6buf = t5buf + NCHW;
  float* t1buf = t6buf + NCHW;                // N*HW
  float* weff  = t1buf + (size_t)N_ * HW_;    // 128*15
  float* Mpart = weff + C_ * 15;              // KSPLIT*N*C*C
  float* Mmat  = Mpart + (size_t)KSPLIT * N_ * C_ * C_;   // N*C*C

  const float scale = 1.0f / (sqrtf((float)C_) * sqrtf((float)HW_));

  // 1) t5
  k_t5<<<dim3((NCHW + 255) / 256), dim3(256), 0, stream>>>(x, p5, t5buf);
  // 2) channel max + effective conv weights
  k_t1<<<dim3((N_ * HW_ + 255) / 256), dim3(256), 0, stream>>>(x, t1buf);
  k_weff<<<dim3((C_ * 15 + 255) / 256), dim3(256), 0, stream>>>(w10, w8, weff);
  // 3) t6 = w6 @ t5   (A shared across batches: strideA = 0)
  gemm_nn_wmma<<<dim3(HW_ / 64, C_ / 16, N_), dim3(32), 0, stream>>>(
      w6, 0, C_, t5buf, NCHW / N_, HW_, t6buf, NCHW / N_, HW_, C_, 1.0f);
  // 4) t12 = max(x, rolled 15-tap conv of t1)  (overwrites t5 buffer)
  k_t12<<<dim3((NCHW + 255) / 256), dim3(256), 0, stream>>>(x, t1buf, weff, t5buf);
  // 5) Mpart[ks][n] = t12[n] @ t6[n]^T  (split-K), then reduce
  gemm_nt_wmma<<<dim3(C_ / 16, C_ / 16, N_ * KSPLIT), dim3(32), 0, stream>>>(
      t5buf, NCHW / N_, HW_, t6buf, NCHW / N_, HW_, Mpart, HW_);
  k_reduceM<<<dim3((N_ * C_ * C_ + 255) / 256), dim3(256), 0, stream>>>(Mpart, Mmat);
  // 6) out = scale * M @ x
  gemm_nn_wmma<<<dim3(HW_ / 64, C_ / 16, N_), dim3(32), 0, stream>>>(
      Mmat, (size_t)C_ * C_, C_, x, NCHW / N_, HW_, out, NCHW / N_, HW_, C_, scale);
}